// DentateGyrus_420906795394
// MI455X (gfx1250) — compile-verified
//
#include <hip/hip_runtime.h>

#define COORD_DIM   64
#define DG_DIM      512
#define K_ACTIVE    20
#define BATCH       131072
#define EPS         1e-5f
#define ROWS_PER_WG 64
#define LDS_STRIDE  516   // floats; 516 % 64 = 4 -> row+8 shifts banks by 32 (conflict-free WMMA writeback)

typedef __attribute__((ext_vector_type(2))) float v2f;
typedef __attribute__((ext_vector_type(8))) float v8f;

// Order-preserving float -> uint key (total order matches float <)
__device__ __forceinline__ unsigned f2key(float f) {
    unsigned u = __float_as_uint(f);
    return u ^ (unsigned)(((int)u >> 31) | 0x80000000);
}

extern "C" __global__ __launch_bounds__(256)
void dg_fused_kernel(const float* __restrict__ x,     // [B, 64]
                     const float* __restrict__ W,     // [64, 512]
                     const float* __restrict__ bias,  // [512]
                     const float* __restrict__ gamma, // [512]
                     const float* __restrict__ beta,  // [512]
                     float* __restrict__ out)         // [B, 512]
{
    extern __shared__ float hbuf[]; // ROWS_PER_WG * LDS_STRIDE floats

    const int tid    = threadIdx.x;
    const int wave   = tid >> 5;
    const int lane   = tid & 31;
    const int l16    = lane & 15;
    const int lhalf  = lane >> 4;       // 0: lanes 0-15, 1: lanes 16-31
    const int wgRow0 = blockIdx.x * ROWS_PER_WG;

    // ---------------- Phase A: x@W via V_WMMA_F32_16X16X4_F32, +bias, ReLU -> LDS ----------------
    // Wave w covers column tiles n0 = (w*4 + c)*16, c = 0..3 (64 columns per wave).
    for (int rt = 0; rt < ROWS_PER_WG / 16; ++rt) {
        const int row0 = wgRow0 + rt * 16;

        // A fragments for the 16x64 row tile: 16 k-steps of 16x4.
        // Layout: lane<16 -> row=l16, K = 4k+{0,1}; lane>=16 -> row=l16, K = 4k+{2,3}.
        // Kept RT-cached: all 8 waves of the WG reuse the same x tiles.
        v2f a[16];
        const float* Ap = x + (size_t)(row0 + l16) * COORD_DIM + 2 * lhalf;
        #pragma unroll
        for (int kk = 0; kk < 16; ++kk)
            a[kk] = *(const v2f*)(Ap + 4 * kk);

        #pragma unroll
        for (int c = 0; c < 4; ++c) {
            const int n0 = (wave * 4 + c) * 16;
            v8f acc = {};
            #pragma unroll
            for (int kk = 0; kk < 16; ++kk) {
                // B fragment 4x16: lane<16 -> K = 4k+{0,1}, N = n0+l16 ; lane>=16 -> K = 4k+{2,3}.
                // Compiler hoists these W fragments into persistent VGPRs across row tiles.
                const float* Wp = W + (size_t)(4 * kk + 2 * lhalf) * DG_DIM + n0 + l16;
                v2f bf;
                bf.x = Wp[0];
                bf.y = Wp[DG_DIM];
                acc = __builtin_amdgcn_wmma_f32_16x16x4_f32(
                        false, a[kk], false, bf, (short)0, acc, false, false);
            }
            // C layout: VGPR m -> row rt*16 + lhalf*8 + m, col n0 + l16
            const float bv    = bias[n0 + l16];
            const int   rbase = rt * 16 + lhalf * 8;
            const int   col   = n0 + l16;
            #pragma unroll
            for (int m = 0; m < 8; ++m) {
                float hv = acc[m] + bv;
                hv = fmaxf(hv, 0.0f);
                hbuf[(rbase + m) * LDS_STRIDE + col] = hv;
            }
        }
    }
    __syncthreads();

    // ---------------- Phase B: LayerNorm + top-K threshold + sparse dense write ----------------
    // Each wave handles 8 rows; lane i covers columns {lane, lane+32, ..., lane+480}.
    for (int j = 0; j < ROWS_PER_WG / 8; ++j) {
        const int r    = wave * (ROWS_PER_WG / 8) + j;
        const int grow = wgRow0 + r;

        float vals[16];
        float sum = 0.0f, sumsq = 0.0f;
        #pragma unroll
        for (int i = 0; i < 16; ++i) {
            const float hv = hbuf[r * LDS_STRIDE + lane + 32 * i];
            vals[i] = hv;
            sum   += hv;
            sumsq += hv * hv;
        }
        #pragma unroll
        for (int off = 16; off >= 1; off >>= 1) {
            sum   += __shfl_xor(sum,   off, 32);
            sumsq += __shfl_xor(sumsq, off, 32);
        }
        const float mu   = sum * (1.0f / DG_DIM);
        const float var  = sumsq * (1.0f / DG_DIM) - mu * mu;
        const float rinv = rsqrtf(var + EPS);

        unsigned keys[16];
        #pragma unroll
        for (int i = 0; i < 16; ++i) {
            const int col = lane + 32 * i;
            float p = (vals[i] - mu) * rinv * gamma[col] + beta[col];
            vals[i] = p;
            keys[i] = f2key(p);
        }

        // Extract max K_ACTIVE times; thr ends as the key of the 20th largest element.
        unsigned thr = 0;
        for (int t = 0; t < K_ACTIVE; ++t) {
            unsigned lmax = 0;
            #pragma unroll
            for (int i = 0; i < 16; ++i) lmax = (keys[i] > lmax) ? keys[i] : lmax;
            unsigned gmax = lmax;
            #pragma unroll
            for (int off = 16; off >= 1; off >>= 1) {
                const unsigned o = __shfl_xor(gmax, off, 32);
                gmax = (o > gmax) ? o : gmax;
            }
            if (lmax == gmax) {          // owning lane(s) remove one matching element
                bool removed = false;
                #pragma unroll
                for (int i = 0; i < 16; ++i) {
                    if (!removed && keys[i] == gmax) { keys[i] = 0; removed = true; }
                }
            }
            thr = gmax;
        }

        // Write-once, never-read 268 MB stream -> non-temporal stores (TH=NT),
        // keeps L2 free for W/gamma/beta and x-tile reuse.
        float* orow = out + (size_t)grow * DG_DIM;
        #pragma unroll
        for (int i = 0; i < 16; ++i) {
            const int col = lane + 32 * i;
            const float p = vals[i];
            const float o = (f2key(p) >= thr) ? p : 0.0f;
            __builtin_nontemporal_store(o, orow + col);
        }
    }
}

extern "C" void kernel_launch(void* const* d_in, const int* in_sizes, int n_in,
                              void* d_out, int out_size, void* d_ws, size_t ws_size,
                              hipStream_t stream) {
    (void)in_sizes; (void)n_in; (void)out_size; (void)d_ws; (void)ws_size;
    const float* x     = (const float*)d_in[0];
    const float* W     = (const float*)d_in[1];
    const float* bias  = (const float*)d_in[2];
    const float* gamma = (const float*)d_in[3];
    const float* beta  = (const float*)d_in[4];
    float* out = (float*)d_out;

    const dim3 grid(BATCH / ROWS_PER_WG);
    const dim3 block(256);
    const size_t ldsBytes = (size_t)ROWS_PER_WG * LDS_STRIDE * sizeof(float);
    hipLaunchKernelGGL(dg_fused_kernel, grid, block, ldsBytes, stream,
                       x, W, bias, gamma, beta, out);
}